// TrueDEQSpring_59691455480097
// MI455X (gfx1250) — compile-verified
//
#include <hip/hip_runtime.h>
#include <math.h>

#define HID 64
#define MAX_ITER 80
#define TOL 1e-8f

typedef __attribute__((ext_vector_type(2))) float v2f;
typedef __attribute__((ext_vector_type(8))) float v8f;

// ws layout (floats):
//   [0]        = scale   (written as uint bits via atomicMax; |w| >= 0 so bit order == float order)
//   [1]        = F0      = sum sigmoid(b1)*w1*softplus(w2_raw)
//   [2..65]    = w1
//   [66..129]  = b1
//   [130..193] = vF = w1*w2
//   [194..257] = vK = w1*w1*w2

__global__ void tds_init_ws(unsigned* wsb) {
    if (threadIdx.x == 0) wsb[0] = 0x3F800000u;  // 1.0f
}

__global__ void tds_scale_reduce(const float* __restrict__ w, unsigned* wsb, int n) {
    __shared__ unsigned sm[256];
    unsigned m = 0u;
    for (int i = blockIdx.x * blockDim.x + threadIdx.x; i < n; i += gridDim.x * blockDim.x)
        m = max(m, __float_as_uint(fabsf(w[i])));
    sm[threadIdx.x] = m;
    __syncthreads();
    for (int s = 128; s > 0; s >>= 1) {
        if ((int)threadIdx.x < s) sm[threadIdx.x] = max(sm[threadIdx.x], sm[threadIdx.x + s]);
        __syncthreads();
    }
    if (threadIdx.x == 0) atomicMax(wsb, sm[0]);
}

__global__ void tds_precompute(const float* __restrict__ w1, const float* __restrict__ b1,
                               const float* __restrict__ w2r, float* __restrict__ ws) {
    int h = threadIdx.x;  // 64 threads
    float a = w1[h], b = b1[h], x = w2r[h];
    float w2 = (x > 20.f) ? x : log1pf(__expf(x));  // softplus, stable
    float vF = a * w2;
    ws[2 + h]   = a;
    ws[66 + h]  = b;
    ws[130 + h] = vF;
    ws[194 + h] = a * a * w2;
    __shared__ float red[64];
    red[h] = vF / (1.f + __expf(-b));  // sigmoid(b1)*w1*w2 (full-precision, runs once)
    __syncthreads();
    for (int s = 32; s > 0; s >>= 1) {
        if (h < s) red[h] += red[h + s];
        __syncthreads();
    }
    if (h == 0) ws[1] = red[0];
}

// Fast sigmoid: v_exp_f32 + v_rcp_f32 (both TRANS ops, co-execute with VALU).
// ~1e-7 relative error, same order as v_exp_f32 itself; avoids the ~9-op IEEE
// division sequence on the transcendental-bound critical path.
__device__ __forceinline__ float fast_sigmoid(float x) {
    return __builtin_amdgcn_rcpf(1.f + __expf(-x));
}

// One wave owns 16 elements. Lane L handles element (L&15); lanes L and L+16
// duplicate the scalar state of that element and supply hidden units
// {4c, 4c+1} (L<16) / {4c+2, 4c+3} (L>=16) of each K=4 chunk as the f32 B
// fragment of V_WMMA_F32_16X16X4_F32. A = coefficient chunk broadcast over
// rows, so D[:,N] = element N's partial dot and every lane reads its own
// result from accumulator VGPR 0. Full fp32 precision end-to-end.
__global__ __launch_bounds__(256) void tds_solver(const float* __restrict__ w,
                                                  const float* __restrict__ ws,
                                                  float* __restrict__ out, int n) {
    __shared__ float sw1[HID], sb1[HID], svF[HID], svK[HID];
    if (threadIdx.x < HID) {
        int t = threadIdx.x;
        sw1[t] = ws[2 + t];
        sb1[t] = ws[66 + t];
        svF[t] = ws[130 + t];
        svK[t] = ws[194 + t];
    }
    __syncthreads();

    const float scale = ws[0];
    const float F0    = ws[1];

    const int lane   = threadIdx.x & 31;
    const int waveId = (int)((blockIdx.x * blockDim.x + threadIdx.x) >> 5);
    const int elem   = waveId * 16 + (lane & 15);
    const int koff   = (lane & 16) >> 3;  // 0 for lanes 0-15, 2 for lanes 16-31
    const bool valid = elem < n;
    const float we   = valid ? w[elem] : 0.f;

    float eps = fminf(fmaxf(we / scale, 0.f), 2.f);
    float G, K;

    // Fused force+stiffness evaluation at e (shares the sigmoid batch).
    auto evalGK = [&](float e, float& Go, float& Ko) {
        v8f accF = {};
        v8f accK = {};
        for (int c = 0; c < 16; ++c) {
            const int h = (c << 2) + koff;
            float sa = fast_sigmoid(e * sw1[h] + sb1[h]);
            float sb = fast_sigmoid(e * sw1[h + 1] + sb1[h + 1]);
            v2f bS; bS.x = sa; bS.y = sb;
            v2f aF; aF.x = svF[h]; aF.y = svF[h + 1];
            accF = __builtin_amdgcn_wmma_f32_16x16x4_f32(false, aF, false, bS,
                                                         (short)0, accF, false, false);
            v2f bD; bD.x = sa * (1.f - sa); bD.y = sb * (1.f - sb);
            v2f aK; aK.x = svK[h]; aK.y = svK[h + 1];
            accK = __builtin_amdgcn_wmma_f32_16x16x4_f32(false, aK, false, bD,
                                                         (short)0, accK, false, false);
        }
        Go = accF[0] - F0 - we;
        Ko = accK[0];
    };

    evalGK(eps, G, K);

    const float damps[5] = {1.f, 0.5f, 0.25f, 0.1f, 0.05f};

    for (int it = 0; it < MAX_ITER; ++it) {
        // wave-uniform exit -> EXEC is all-ones around every WMMA
        if (!__any(fabsf(G) >= TOL)) break;

        float step = G / fmaxf(K, 1e-8f);
        float et[5];
#pragma unroll
        for (int t = 0; t < 5; ++t)
            et[t] = fminf(fmaxf(eps - damps[t] * step, 0.f), 2.f);

        // 5 damped-trial force evaluations, one WMMA accumulator per trial.
        v8f acc[5] = {};
        for (int c = 0; c < 16; ++c) {
            const int h = (c << 2) + koff;
            float w1a = sw1[h], w1b = sw1[h + 1];
            float b1a = sb1[h], b1b = sb1[h + 1];
            v2f aF; aF.x = svF[h]; aF.y = svF[h + 1];
#pragma unroll
            for (int t = 0; t < 5; ++t) {
                v2f bS;
                bS.x = fast_sigmoid(et[t] * w1a + b1a);
                bS.y = fast_sigmoid(et[t] * w1b + b1b);
                acc[t] = __builtin_amdgcn_wmma_f32_16x16x4_f32(false, aF, false, bS,
                                                               (short)0, acc[t], false, false);
            }
        }

        const float thr = fabsf(G) + 1e-12f;
        float eps_new = fminf(fmaxf(eps - 1e-4f * G, 0.f), 2.f);  // gradient fallback
        bool found = false;
#pragma unroll
        for (int t = 0; t < 5; ++t) {  // first accepted damping (damps[0]=1.0 first)
            float Gt = acc[t][0] - F0 - we;
            if (!found && fabsf(Gt) <= thr) { eps_new = et[t]; found = true; }
        }

        eps = eps_new;
        evalGK(eps, G, K);  // G_new; K reused at same eps next iteration (identical value)
    }

    if (valid && lane < 16) out[elem] = eps;
}

extern "C" void kernel_launch(void* const* d_in, const int* in_sizes, int n_in,
                              void* d_out, int out_size, void* d_ws, size_t ws_size,
                              hipStream_t stream) {
    const float* w   = (const float*)d_in[0];
    const float* w1  = (const float*)d_in[1];
    const float* b1  = (const float*)d_in[2];
    const float* w2r = (const float*)d_in[3];
    float* out = (float*)d_out;
    float* ws  = (float*)d_ws;
    const int n = in_sizes[0];

    tds_init_ws<<<1, 1, 0, stream>>>((unsigned*)ws);
    int rblocks = (n + 255) / 256;
    if (rblocks > 256) rblocks = 256;
    tds_scale_reduce<<<rblocks, 256, 0, stream>>>(w, (unsigned*)ws, n);
    tds_precompute<<<1, HID, 0, stream>>>(w1, b1, w2r, ws);

    const int waves  = (n + 15) / 16;
    const int blocks = (waves + 7) / 8;  // 8 waves (256 threads) per block
    tds_solver<<<blocks, 256, 0, stream>>>(w, ws, out, n);
}